// Beamform_1649267442279
// MI455X (gfx1250) — compile-verified
//
#include <hip/hip_runtime.h>

typedef float v2f __attribute__((ext_vector_type(2)));
typedef float v8f __attribute__((ext_vector_type(8)));

#define NBATCH_PER_CH 500000
#define NBATCH        2000000
#define NGROUPS       666666        // groups of 3 batches: covers batches [0, 1999998)
#define TAILB         (NGROUPS * 3) // 1999998 ; 2 tail batches

// out[b, j] : j<5 -> real_k=j, j>=5 -> imag_k=j-5
// out_real[b,k] = sum_r ( w_re[r]*x_re[b,r,k] - w_im[r]*x_im[b,r,k] )
// out_imag[b,k] = sum_r ( w_im[r]*x_re[b,r,k] + w_re[r]*x_im[b,r,k] )
// x_re[b,r,k] = ch_c[ (b%500000)*40 + (r*5+k)*2 ],  x_im = +1 (interleaved)

__global__ __launch_bounds__(256)
void beamform_wmma_f32(const float* __restrict__ in0,
                       const float* __restrict__ in1,
                       const float* __restrict__ in2,
                       const float* __restrict__ in3,
                       const float* __restrict__ bf,
                       float* __restrict__ out,
                       int nwaves)
{
    const int lane = threadIdx.x & 31;
    const int wave = blockIdx.x * (blockDim.x >> 5) + (threadIdx.x >> 5);
    const int n    = lane & 15;   // WMMA column index (lanes 16-31 mirror columns)
    const int half = lane >> 4;   // 0 -> K rows 0,1 ; 1 -> K rows 2,3
    const int r0   = half << 1;

    // weights: bf = [w_re0, w_im0, w_re1, w_im1, ...] (uniform load)
    const float w0 = bf[0], w1 = bf[1], w2 = bf[2], w3 = bf[3];
    const float w4 = bf[4], w5 = bf[5], w6 = bf[6], w7 = bf[7];

    // A fragments (16x4 f32 layout: lanes 0-15 -> VGPR0=K0, VGPR1=K1; lanes 16-31 -> K2,K3)
    const float wreK0 = half ? w4 : w0;   // w_re[r0]
    const float wimK0 = half ? w5 : w1;   // w_im[r0]
    const float wreK1 = half ? w6 : w2;   // w_re[r0+1]
    const float wimK1 = half ? w7 : w3;   // w_im[r0+1]
    v2f a_re, a_im;
    a_re.x = (n == 0) ?  wreK0 : (n == 1) ? wimK0 : 0.0f;
    a_re.y = (n == 0) ?  wreK1 : (n == 1) ? wimK1 : 0.0f;
    a_im.x = (n == 0) ? -wimK0 : (n == 1) ? wreK0 : 0.0f;
    a_im.y = (n == 0) ? -wimK1 : (n == 1) ? wreK1 : 0.0f;

    // column decode: column n -> (batch-in-group g, output pos k); clamp n==15 (unused col)
    int g = n / 5;     g = (g > 2) ? 2 : g;
    int k = n - g * 5; k = (k > 4) ? 4 : k;
    const int fragoff = (r0 * 5 + k) * 2; // float offset of x_re[r0, k] within 40-float batch

    for (int grp = wave; grp < NGROUPS; grp += nwaves) {
        const int b   = grp * 3 + g;
        const int c   = b / NBATCH_PER_CH;
        const int off = b - c * NBATCH_PER_CH;
        const float* ch = (c & 2) ? ((c & 1) ? in3 : in2)
                                  : ((c & 1) ? in1 : in0);
        const float* p = ch + (size_t)off * 40 + fragoff;

        // (x_re, x_im) for K=r0 and K=r0+1 : two NT b64 loads, 8B apart rows are +10 floats
        v2f l0 = __builtin_nontemporal_load((const v2f*)p);
        v2f l1 = __builtin_nontemporal_load((const v2f*)(p + 10));
        v2f b_re; b_re.x = l0.x; b_re.y = l1.x;  // B 4x16: VGPR0 = K{0|2}, VGPR1 = K{1|3}
        v2f b_im; b_im.x = l0.y; b_im.y = l1.y;

        v8f acc = {};
        acc = __builtin_amdgcn_wmma_f32_16x16x4_f32(false, a_re, false, b_re,
                                                    (short)0, acc, false, false);
        acc = __builtin_amdgcn_wmma_f32_16x16x4_f32(false, a_im, false, b_im,
                                                    (short)0, acc, false, false);

        // D row0 = out_real, row1 = out_imag, valid in lanes 0..14 (columns 0..14)
        if (lane < 15) {
            const int bb = grp * 3 + lane / 5;
            const int kk = lane % 5;
            float* q = out + (size_t)bb * 10 + kk;
            __builtin_nontemporal_store(acc[0], q);      // real at j=kk
            __builtin_nontemporal_store(acc[1], q + 5);  // imag at j=kk+5
        }
    }

    // tail: batches 1999998, 1999999 (channel 3) -> 20 scalar outputs
    const int gtid = blockIdx.x * blockDim.x + threadIdx.x;
    if (gtid < (NBATCH - TAILB) * 10) {
        const int b  = TAILB + gtid / 10;
        const int j  = gtid % 10;
        const int k2 = (j < 5) ? j : j - 5;
        const float* base = in3 + (size_t)(b - 3 * NBATCH_PER_CH) * 40;
        const float wre[4] = {w0, w2, w4, w6};
        const float wim[4] = {w1, w3, w5, w7};
        float re = 0.0f, im = 0.0f;
        #pragma unroll
        for (int r = 0; r < 4; ++r) {
            const float xr = base[(r * 5 + k2) * 2];
            const float xi = base[(r * 5 + k2) * 2 + 1];
            re = __builtin_fmaf(wre[r], xr, re); re = __builtin_fmaf(-wim[r], xi, re);
            im = __builtin_fmaf(wim[r], xr, im); im = __builtin_fmaf( wre[r], xi, im);
        }
        out[(size_t)b * 10 + j] = (j < 5) ? re : im;
    }
}

extern "C" void kernel_launch(void* const* d_in, const int* in_sizes, int n_in,
                              void* d_out, int out_size, void* d_ws, size_t ws_size,
                              hipStream_t stream) {
    const float* in0 = (const float*)d_in[0];
    const float* in1 = (const float*)d_in[1];
    const float* in2 = (const float*)d_in[2];
    const float* in3 = (const float*)d_in[3];
    const float* bf  = (const float*)d_in[4];
    float* out = (float*)d_out;

    const int blocks  = 2048;
    const int threads = 256;                       // 8 wave32 per block
    const int nwaves  = blocks * (threads / 32);   // 16384 persistent waves
    beamform_wmma_f32<<<blocks, threads, 0, stream>>>(in0, in1, in2, in3, bf, out, nwaves);
}